// EncoderEdgeClassify_62508954026584
// MI455X (gfx1250) — compile-verified
//
#include <hip/hip_runtime.h>
#include <hip/hip_bf16.h>
#include <math.h>

#define B_    256
#define S_    43
#define D_    18
#define DP_   20          /* D padded to multiple of 4 */
#define H_    3
#define DK_   6
#define L_    6
#define FF_   2048
#define A_    128
#define NB_   7
#define KR_   44          /* S padded to multiple of 4 */
#define ROWS_ (B_ * S_)   /* 11008 = 688 * 16 */
#define MT_   (ROWS_ / 16)
#define EPS_  1e-6f

/* combined packed chunk: [320 floats W1 | 512 floats W2] per F-chunk */
#define CHUNK_F       832
#define CHUNK_BYTES   (CHUNK_F * 4)              /* 3328 B = 208 x 16 B */
#define WCP_PER_LAYER (128 * CHUNK_F)            /* 106496 floats       */

#define FFN_WAVES 8        /* waves (M-tiles) per FFN block */
#define STG_F     CHUNK_F  /* floats per stage buffer       */

typedef __attribute__((ext_vector_type(2))) float v2f;
typedef __attribute__((ext_vector_type(8))) float v8f;

typedef int v4i_ __attribute__((vector_size(16)));
typedef __attribute__((address_space(3))) v4i_* lds_v4i_p;

static __device__ __forceinline__ v8f wmma_f32_k4(v2f a, v2f b, v8f c) {
    return __builtin_amdgcn_wmma_f32_16x16x4_f32(false, a, false, b, (short)0, c,
                                                 false, false);
}

// --- CDNA5 async global->LDS copy (ASYNCcnt) with guarded fallback ---------
#if __has_builtin(__builtin_amdgcn_global_load_async_to_lds_b128)
#define HAVE_ASYNC_LDS 1
static __device__ __forceinline__ void async_cp16(const void* g, void* l) {
    __builtin_amdgcn_global_load_async_to_lds_b128((v4i_*)g, (lds_v4i_p)l, 0, 0);
}
#else
#define HAVE_ASYNC_LDS 0
static __device__ __forceinline__ void async_cp16(const void* g, void* l) {
    *(float4*)l = *(const float4*)g;   // load + ds_store; ordered by barrier
}
#endif

#if HAVE_ASYNC_LDS
#if __has_builtin(__builtin_amdgcn_s_wait_asynccnt)
#define WAIT_ASYNC(n) __builtin_amdgcn_s_wait_asynccnt(n)
#else
#define WAIT_ASYNC(n) asm volatile("s_wait_asynccnt %0" ::"i"(n) : "memory")
#endif
#else
#define WAIT_ASYNC(n) ((void)0)
#endif

// ---------------------------------------------------------------------------
// Weight pre-packing into the combined per-chunk WMMA B-operand layout:
//   Wcp[layer][fc] = [ 5 k-steps x 32 lanes x v2f (W1) | 8 (kk,tile) x 32 x v2f (W2) ]
// Zero-padding (K 18->20 for W1, N 18->32 for W2) baked in.
// ---------------------------------------------------------------------------
__global__ void pack_w1_kernel(const float* __restrict__ W1all,
                               float* __restrict__ p) {
    int id = blockIdx.x * blockDim.x + threadIdx.x;
    if (id >= L_ * 128 * 5 * 32) return;
    int lane = id & 31;
    int t = id >> 5;
    int s = t % 5; int u = t / 5;
    int fc = u & 127; int layer = u >> 7;
    int k0 = s * 4 + ((lane >> 4) & 1) * 2;
    int n  = fc * 16 + (lane & 15);
    const float* W = W1all + (size_t)layer * D_ * FF_;
    float v0 = (k0 < D_)     ? W[(size_t)k0 * FF_ + n]       : 0.f;
    float v1 = (k0 + 1 < D_) ? W[(size_t)(k0 + 1) * FF_ + n] : 0.f;
    float* dst = p + (size_t)layer * WCP_PER_LAYER + (size_t)fc * CHUNK_F +
                 (size_t)((s * 32) + lane) * 2;
    dst[0] = v0; dst[1] = v1;
}

__global__ void pack_w2_kernel(const float* __restrict__ W2all,
                               float* __restrict__ p) {
    int id = blockIdx.x * blockDim.x + threadIdx.x;
    if (id >= L_ * 128 * 4 * 2 * 32) return;
    int lane = id & 31;
    int tl   = (id >> 5) & 1;
    int kk   = (id >> 6) & 3;
    int fc   = (id >> 8) & 127;
    int layer = id >> 15;
    int krow = fc * 16 + kk * 4 + ((lane >> 4) & 1) * 2;
    int col  = tl * 16 + (lane & 15);
    const float* W = W2all + (size_t)layer * FF_ * D_;
    float v0 = (col < D_) ? W[(size_t)krow * D_ + col]       : 0.f;
    float v1 = (col < D_) ? W[(size_t)(krow + 1) * D_ + col] : 0.f;
    float* dst = p + (size_t)layer * WCP_PER_LAYER + (size_t)fc * CHUNK_F + 320 +
                 (size_t)(((kk * 2 + tl) * 32) + lane) * 2;
    dst[0] = v0; dst[1] = v1;
}

// ll2_W: (18 x 128) -> [nt(8)][s(5)][lane][2]
__global__ void pack_ll2_kernel(const float* __restrict__ W,
                                float* __restrict__ p) {
    int id = blockIdx.x * blockDim.x + threadIdx.x;
    if (id >= 8 * 5 * 32) return;
    int lane = id & 31;
    int s  = (id >> 5) % 5;
    int nt = (id >> 5) / 5;
    int k0 = s * 4 + ((lane >> 4) & 1) * 2;
    int n  = nt * 16 + (lane & 15);
    float v0 = (k0 < D_)     ? W[(size_t)k0 * A_ + n]       : 0.f;
    float v1 = (k0 + 1 < D_) ? W[(size_t)(k0 + 1) * A_ + n] : 0.f;
    p[(size_t)id * 2]     = v0;
    p[(size_t)id * 2 + 1] = v1;
}

// ll_W: (43 x 128) -> [nt(8)][s(11)][lane][2]
__global__ void pack_llw_kernel(const float* __restrict__ W,
                                float* __restrict__ p) {
    int id = blockIdx.x * blockDim.x + threadIdx.x;
    if (id >= 8 * 11 * 32) return;
    int lane = id & 31;
    int s  = (id >> 5) % 11;
    int nt = (id >> 5) / 11;
    int k0 = s * 4 + ((lane >> 4) & 1) * 2;
    int n  = nt * 16 + (lane & 15);
    float v0 = (k0 < S_)     ? W[(size_t)k0 * A_ + n]       : 0.f;
    float v1 = (k0 + 1 < S_) ? W[(size_t)(k0 + 1) * A_ + n] : 0.f;
    p[(size_t)id * 2]     = v0;
    p[(size_t)id * 2 + 1] = v1;
}

// ---------------------------------------------------------------------------
// LayerNorm (ddof=1); ostride > 18 pads the tail with zeros (A-operand ready)
// ---------------------------------------------------------------------------
__global__ void ln_kernel(const float* __restrict__ x,
                          const float* __restrict__ ga,
                          const float* __restrict__ gb,
                          float* __restrict__ y, int rows, int ostride) {
    int row = blockIdx.x * blockDim.x + threadIdx.x;
    if (row >= rows) return;
    const float* xr = x + (size_t)row * D_;
    float v[D_];
    float mu = 0.f;
#pragma unroll
    for (int i = 0; i < D_; ++i) { v[i] = xr[i]; mu += v[i]; }
    mu *= (1.0f / D_);
    float var = 0.f;
#pragma unroll
    for (int i = 0; i < D_; ++i) { float d = v[i] - mu; var += d * d; }
    float sd  = sqrtf(var * (1.0f / (D_ - 1)));
    float inv = 1.0f / (sd + EPS_);
    float* yr = y + (size_t)row * ostride;
#pragma unroll
    for (int i = 0; i < D_; ++i) yr[i] = ga[i] * (v[i] - mu) * inv + gb[i];
    for (int i = D_; i < ostride; ++i) yr[i] = 0.f;
}

// Pad x (stride 18) -> xp (stride 20, zero tail)
__global__ void pad_x_kernel(const float* __restrict__ x, float* __restrict__ xp) {
    int id = blockIdx.x * blockDim.x + threadIdx.x;
    if (id >= ROWS_ * DP_) return;
    int col = id % DP_, row = id / DP_;
    xp[id] = (col < D_) ? x[(size_t)row * D_ + col] : 0.f;
}

// ---------------------------------------------------------------------------
// QKV projection
// ---------------------------------------------------------------------------
__global__ void qkv_kernel(const float* __restrict__ x2,
                           const float* __restrict__ Wq, const float* __restrict__ bq,
                           const float* __restrict__ Wk, const float* __restrict__ bk,
                           const float* __restrict__ Wv, const float* __restrict__ bv,
                           float* __restrict__ q, float* __restrict__ k,
                           float* __restrict__ v) {
    int id = blockIdx.x * blockDim.x + threadIdx.x;
    if (id >= ROWS_ * D_) return;
    int row = id / D_, col = id % D_;
    const float* xr = x2 + (size_t)row * D_;
    float aq = bq[col], ak = bk[col], av = bv[col];
#pragma unroll
    for (int t = 0; t < D_; ++t) {
        float xv = xr[t];
        aq += xv * Wq[t * D_ + col];
        ak += xv * Wk[t * D_ + col];
        av += xv * Wv[t * D_ + col];
    }
    q[id] = aq; k[id] = ak; v[id] = av;
}

// ---------------------------------------------------------------------------
// Attention core: one thread per (b,h,query); two-pass softmax over 43 keys
// ---------------------------------------------------------------------------
__global__ void attn_kernel(const float* __restrict__ q,
                            const float* __restrict__ k,
                            const float* __restrict__ v,
                            const int* __restrict__ mask,
                            float* __restrict__ o) {
    int id = blockIdx.x * blockDim.x + threadIdx.x;
    if (id >= B_ * H_ * S_) return;
    int qi = id % S_;
    int h  = (id / S_) % H_;
    int b  = id / (S_ * H_);
    const float scale = rsqrtf((float)DK_);
    float qv[DK_];
#pragma unroll
    for (int d = 0; d < DK_; ++d)
        qv[d] = q[((size_t)(b * S_ + qi)) * D_ + h * DK_ + d];

    float mx = -1e30f;
    for (int j = 0; j < S_; ++j) {
        const float* kr = k + ((size_t)(b * S_ + j)) * D_ + h * DK_;
        float s = 0.f;
#pragma unroll
        for (int d = 0; d < DK_; ++d) s += qv[d] * kr[d];
        s *= scale;
        if (mask[b * S_ + j] == 0) s = -1e9f;
        mx = fmaxf(mx, s);
    }
    float den = 0.f;
    float acc[DK_] = {0.f, 0.f, 0.f, 0.f, 0.f, 0.f};
    for (int j = 0; j < S_; ++j) {
        const float* kr = k + ((size_t)(b * S_ + j)) * D_ + h * DK_;
        float s = 0.f;
#pragma unroll
        for (int d = 0; d < DK_; ++d) s += qv[d] * kr[d];
        s *= scale;
        if (mask[b * S_ + j] == 0) s = -1e9f;
        float w = expf(s - mx);
        den += w;
        const float* vr = v + ((size_t)(b * S_ + j)) * D_ + h * DK_;
#pragma unroll
        for (int d = 0; d < DK_; ++d) acc[d] += w * vr[d];
    }
    float inv = 1.0f / den;
#pragma unroll
    for (int d = 0; d < DK_; ++d)
        o[((size_t)(b * S_ + qi)) * D_ + h * DK_ + d] = acc[d] * inv;
}

// ---------------------------------------------------------------------------
// Output projection + residual
// ---------------------------------------------------------------------------
__global__ void proj_add_kernel(const float* __restrict__ o,
                                const float* __restrict__ Wo,
                                const float* __restrict__ bo,
                                float* __restrict__ x) {
    int id = blockIdx.x * blockDim.x + threadIdx.x;
    if (id >= ROWS_ * D_) return;
    int row = id / D_, col = id % D_;
    const float* orow = o + (size_t)row * D_;
    float acc = bo[col];
#pragma unroll
    for (int t = 0; t < D_; ++t) acc += orow[t] * Wo[t * D_ + col];
    x[id] += acc;
}

// ---------------------------------------------------------------------------
// Fused FFN, WMMA f32 16x16x4. Block = 8 waves (one 16-row M-tile each),
// combined weight chunks double-buffered into LDS via async global->LDS
// copies (branchless staging: one b128 per thread, lanes >207 clamped).
// ---------------------------------------------------------------------------
__global__ __launch_bounds__(256) void ffn_kernel(const float* __restrict__ x2p,
                                                  const float* __restrict__ Wcp,
                                                  const float* __restrict__ b1,
                                                  const float* __restrict__ b2,
                                                  float* __restrict__ x) {
    __shared__ __align__(16) float stage[2][STG_F];   // [0..319]=W1 chunk, [320..831]=W2 chunk
    __shared__ __align__(16) float hlds[FFN_WAVES][256];
    const int tid   = threadIdx.x;
    const int wave  = tid >> 5;
    const int lane  = tid & 31;
    const int lhalf = lane >> 4;
    const int l15   = lane & 15;
    const int tile  = blockIdx.x * FFN_WAVES + wave;
    const int row0  = tile * 16;
    float* hw = hlds[wave];

    // A-operand (x2 tile), K padded in x2p (stride 20)
    v2f a1[5];
#pragma unroll
    for (int s = 0; s < 5; ++s)
        a1[s] = *(const v2f*)&x2p[(size_t)(row0 + l15) * DP_ + s * 4 + lhalf * 2];

    // branchless cooperative stage of one combined chunk (3328 B = 208 x b128)
    const int ct = tid < 207 ? tid : 207;   // lanes 208..255 duplicate slot 207
    auto stage_chunk = [&](int fc, int buf) {
        const char* src = (const char*)Wcp + (size_t)fc * CHUNK_BYTES + (size_t)ct * 16;
        async_cp16(src, &stage[buf][ct * 4]);
    };

    v8f y0 = {};
    v8f y1 = {};

    stage_chunk(0, 0);
    WAIT_ASYNC(0);
    __syncthreads();

    for (int fc = 0; fc < FF_ / 16; ++fc) {
        const int cur = fc & 1;
        if (fc + 1 < FF_ / 16) {
            stage_chunk(fc + 1, cur ^ 1);
            WAIT_ASYNC(1);            // current buffer's stage complete
        } else {
            WAIT_ASYNC(0);
        }
        __syncthreads();              // all waves' parts of stage[cur] visible

        const float* S = stage[cur];
        // GEMM1: h = x2_tile @ W1[:,chunk]
        v8f h = {};
#pragma unroll
        for (int s = 0; s < 5; ++s) {
            v2f bb = *(const v2f*)&S[(s * 32 + lane) * 2];
            h = wmma_f32_k4(a1[s], bb, h);
        }
        float bias = b1[fc * 16 + l15];
#pragma unroll
        for (int r = 0; r < 8; ++r) {
            float t = h[r] + bias;
            h[r] = t > 0.f ? t : 0.f;
        }
        // C-layout -> A-layout via per-wave LDS
#pragma unroll
        for (int r = 0; r < 8; ++r)
            hw[(r + lhalf * 8) * 16 + l15] = h[r];
        __syncthreads();
        // GEMM2: y += h @ W2[chunk,:]
#pragma unroll
        for (int kk = 0; kk < 4; ++kk) {
            v2f a2 = *(const v2f*)&hw[l15 * 16 + kk * 4 + lhalf * 2];
            v2f b0  = *(const v2f*)&S[320 + ((kk * 2 + 0) * 32 + lane) * 2];
            v2f b1v = *(const v2f*)&S[320 + ((kk * 2 + 1) * 32 + lane) * 2];
            y0 = wmma_f32_k4(a2, b0, y0);
            y1 = wmma_f32_k4(a2, b1v, y1);
        }
        __syncthreads();              // done reading stage[cur] before reuse
    }
#pragma unroll
    for (int r = 0; r < 8; ++r) {
        int m = row0 + r + lhalf * 8;
        int n = l15;
        x[(size_t)m * D_ + n] += y0[r] + b2[n];
        int n2 = 16 + l15;
        if (n2 < D_) x[(size_t)m * D_ + n2] += y1[r] + b2[n2];
    }
}

// ---------------------------------------------------------------------------
// red = x @ ll2_W + ll2_b, stored TRANSPOSED per batch: red_t[b][m][k] (K=44)
// ---------------------------------------------------------------------------
__global__ __launch_bounds__(32) void red_kernel(const float* __restrict__ xp,
                                                 const float* __restrict__ Wp,
                                                 const float* __restrict__ bias,
                                                 float* __restrict__ red_t) {
    const int mt = blockIdx.x;
    const int nt = blockIdx.y;
    const int lane = threadIdx.x, lhalf = lane >> 4, l15 = lane & 15;
    v8f c = {};
#pragma unroll
    for (int s = 0; s < 5; ++s) {
        v2f a = *(const v2f*)&xp[(size_t)(mt * 16 + l15) * DP_ + s * 4 + lhalf * 2];
        v2f b = ((const v2f*)Wp)[(size_t)(nt * 5 + s) * 32 + lane];
        c = wmma_f32_k4(a, b, c);
    }
    const int n = nt * 16 + l15;
    float bv = bias[n];
#pragma unroll
    for (int r = 0; r < 8; ++r) {
        int row = mt * 16 + r + lhalf * 8;   // global b*43+s
        int b_  = row / S_;
        int sq  = row - b_ * S_;
        red_t[((size_t)b_ * A_ + n) * KR_ + sq] = c[r] + bv;
    }
}

// zero the K=43 pad column of red_t
__global__ void pad_red_kernel(float* __restrict__ red_t) {
    int id = blockIdx.x * blockDim.x + threadIdx.x;
    if (id >= B_ * A_) return;
    red_t[(size_t)id * KR_ + (KR_ - 1)] = 0.f;
}

// ---------------------------------------------------------------------------
// R[b] = red_t[b] @ ll_W + ll_b : (128x44)@(44x128), fully unpredicated
// ---------------------------------------------------------------------------
__global__ __launch_bounds__(32) void R_kernel(const float* __restrict__ red_t,
                                               const float* __restrict__ Wp,
                                               const float* __restrict__ bias,
                                               float* __restrict__ R) {
    const int b  = blockIdx.x;
    const int mt = blockIdx.y;
    const int nt = blockIdx.z;
    const int lane = threadIdx.x, lhalf = lane >> 4, l15 = lane & 15;
    const int m = mt * 16 + l15;
    const int n = nt * 16 + l15;
    v8f c = {};
#pragma unroll
    for (int s = 0; s < 11; ++s) {
        int k0 = s * 4 + lhalf * 2;
        v2f a  = *(const v2f*)&red_t[((size_t)b * A_ + m) * KR_ + k0];
        v2f bb = ((const v2f*)Wp)[(size_t)(nt * 11 + s) * 32 + lane];
        c = wmma_f32_k4(a, bb, c);
    }
    float bv = bias[n];
#pragma unroll
    for (int r = 0; r < 8; ++r)
        R[((size_t)b * A_ + mt * 16 + r + lhalf * 8) * A_ + n] = c[r] + bv;
}

// ---------------------------------------------------------------------------
// P = R @ fl_W[:128], Q = R @ fl_W[128:]
// ---------------------------------------------------------------------------
__global__ void pq_kernel(const float* __restrict__ R,
                          const float* __restrict__ flW,
                          float* __restrict__ P, float* __restrict__ Q) {
    int id = blockIdx.x * blockDim.x + threadIdx.x;
    if (id >= B_ * A_ * NB_) return;
    int c  = id % NB_;
    int bi = id / NB_;
    const float* Rrow = R + (size_t)bi * A_;
    float p = 0.f, qq = 0.f;
    for (int j = 0; j < A_; ++j) {
        float r = Rrow[j];
        p  += r * flW[(size_t)j * NB_ + c];
        qq += r * flW[(size_t)(A_ + j) * NB_ + c];
    }
    P[id] = p; Q[id] = qq;
}

// ---------------------------------------------------------------------------
// out[b,i,j,c] = P[b,i,c] + Q[b,j,c] + fl_b[c]
// ---------------------------------------------------------------------------
__global__ void out_kernel(const float* __restrict__ P,
                           const float* __restrict__ Q,
                           const float* __restrict__ flb,
                           float* __restrict__ out) {
    long long id = (long long)blockIdx.x * blockDim.x + threadIdx.x;
    const long long total = (long long)B_ * A_ * A_ * NB_;
    if (id >= total) return;
    int c    = (int)(id % NB_);
    int j    = (int)((id / NB_) % A_);
    long long bi = id / ((long long)NB_ * A_);
    long long b  = bi / A_;
    out[id] = P[bi * NB_ + c] + Q[(b * A_ + j) * NB_ + c] + flb[c];
}

// ---------------------------------------------------------------------------
extern "C" void kernel_launch(void* const* d_in, const int* in_sizes, int n_in,
                              void* d_out, int out_size, void* d_ws, size_t ws_size,
                              hipStream_t stream) {
    (void)in_sizes; (void)n_in; (void)out_size; (void)ws_size;
    const float* src  = (const float*)d_in[0];
    const int*   mask = (const int*)d_in[1];
    const float* Wq   = (const float*)d_in[3];
    const float* bq   = (const float*)d_in[4];
    const float* Wk   = (const float*)d_in[5];
    const float* bk   = (const float*)d_in[6];
    const float* Wv   = (const float*)d_in[7];
    const float* bv   = (const float*)d_in[8];
    const float* Wo   = (const float*)d_in[9];
    const float* bo   = (const float*)d_in[10];
    const float* ln1a = (const float*)d_in[11];
    const float* ln1b = (const float*)d_in[12];
    const float* ln2a = (const float*)d_in[13];
    const float* ln2b = (const float*)d_in[14];
    const float* fW1  = (const float*)d_in[15];
    const float* fb1  = (const float*)d_in[16];
    const float* fW2  = (const float*)d_in[17];
    const float* fb2  = (const float*)d_in[18];
    const float* ll2W = (const float*)d_in[19];
    const float* ll2b = (const float*)d_in[20];
    const float* llW  = (const float*)d_in[21];
    const float* llb  = (const float*)d_in[22];
    const float* flW  = (const float*)d_in[23];
    const float* flb  = (const float*)d_in[24];

    float* ws = (float*)d_ws;
    size_t off = 0;
    float* xb    = ws + off; off += (size_t)ROWS_ * D_;
    float* x2b   = ws + off; off += (size_t)ROWS_ * D_;     // LN1 out (stride 18)
    float* x2p   = ws + off; off += (size_t)ROWS_ * DP_;    // LN2 out (stride 20)
    float* xp    = ws + off; off += (size_t)ROWS_ * DP_;    // padded x for red
    float* qb    = ws + off; off += (size_t)ROWS_ * D_;
    float* kb    = ws + off; off += (size_t)ROWS_ * D_;
    float* vb    = ws + off; off += (size_t)ROWS_ * D_;
    float* ob    = ws + off; off += (size_t)ROWS_ * D_;
    float* Wcp   = ws + off; off += (size_t)L_ * WCP_PER_LAYER;
    float* ll2Wp = ws + off; off += (size_t)8 * 5 * 32 * 2;
    float* llWp  = ws + off; off += (size_t)8 * 11 * 32 * 2;
    float* redt  = ws + off; off += (size_t)B_ * A_ * KR_;
    float* Rb    = ws + off; off += (size_t)B_ * A_ * A_;
    float* Pb    = ws + off; off += (size_t)B_ * A_ * NB_;
    float* Qb    = ws + off; off += (size_t)B_ * A_ * NB_;

    const int TB = 256;
    // --- packing (rerun every call: no cached state allowed) ---
    pack_w1_kernel<<<(L_ * 128 * 5 * 32 + TB - 1) / TB, TB, 0, stream>>>(fW1, Wcp);
    pack_w2_kernel<<<(L_ * 128 * 4 * 2 * 32 + TB - 1) / TB, TB, 0, stream>>>(fW2, Wcp);
    pack_ll2_kernel<<<(8 * 5 * 32 + TB - 1) / TB, TB, 0, stream>>>(ll2W, ll2Wp);
    pack_llw_kernel<<<(8 * 11 * 32 + TB - 1) / TB, TB, 0, stream>>>(llW, llWp);

    (void)hipMemcpyAsync(xb, src, (size_t)ROWS_ * D_ * sizeof(float),
                         hipMemcpyDeviceToDevice, stream);

    const int gRows = (ROWS_ + TB - 1) / TB;
    const int gElem = (ROWS_ * D_ + TB - 1) / TB;
    const int gAttn = (B_ * H_ * S_ + TB - 1) / TB;

    for (int i = 0; i < L_; ++i) {
        const float* Wqi = Wq + (size_t)i * D_ * D_;
        const float* Wki = Wk + (size_t)i * D_ * D_;
        const float* Wvi = Wv + (size_t)i * D_ * D_;
        const float* Woi = Wo + (size_t)i * D_ * D_;
        ln_kernel<<<gRows, TB, 0, stream>>>(xb, ln1a + i * D_, ln1b + i * D_,
                                            x2b, ROWS_, D_);
        qkv_kernel<<<gElem, TB, 0, stream>>>(x2b, Wqi, bq + i * D_, Wki, bk + i * D_,
                                             Wvi, bv + i * D_, qb, kb, vb);
        attn_kernel<<<gAttn, TB, 0, stream>>>(qb, kb, vb, mask, ob);
        proj_add_kernel<<<gElem, TB, 0, stream>>>(ob, Woi, bo + i * D_, xb);
        ln_kernel<<<gRows, TB, 0, stream>>>(xb, ln2a + i * D_, ln2b + i * D_,
                                            x2p, ROWS_, DP_);
        ffn_kernel<<<MT_ / FFN_WAVES, FFN_WAVES * 32, 0, stream>>>(
            x2p,
            Wcp + (size_t)i * WCP_PER_LAYER,
            fb1 + (size_t)i * FF_,
            fb2 + (size_t)i * D_,
            xb);
    }

    pad_x_kernel<<<(ROWS_ * DP_ + TB - 1) / TB, TB, 0, stream>>>(xb, xp);
    red_kernel<<<dim3(MT_, A_ / 16), 32, 0, stream>>>(xp, ll2Wp, ll2b, redt);
    pad_red_kernel<<<(B_ * A_ + TB - 1) / TB, TB, 0, stream>>>(redt);
    R_kernel<<<dim3(B_, A_ / 16, A_ / 16), 32, 0, stream>>>(redt, llWp, llb, Rb);
    pq_kernel<<<(B_ * A_ * NB_ + TB - 1) / TB, TB, 0, stream>>>(Rb, flW, Pb, Qb);
    const long long total = (long long)B_ * A_ * A_ * NB_;
    out_kernel<<<(int)((total + TB - 1) / TB), TB, 0, stream>>>(Pb, Qb, flb,
                                                                (float*)d_out);
}